// PointPillarScatter_41034117546255
// MI455X (gfx1250) — compile-verified
//
#include <hip/hip_runtime.h>
#include <stdint.h>

// Problem constants (match reference shapes)
#define BATCH   2
#define GIDS    8
#define CFEAT   64
#define NXC     432
#define NYC     496
#define NCELLS  (NXC * NYC)          // 214272  (divisible by 4 and by 256)
#define NPLANES (BATCH * GIDS)       // 16
#define NROWS   (NPLANES * NCELLS)   // 3,428,352

typedef float v4f __attribute__((ext_vector_type(4)));
typedef int   v4i __attribute__((ext_vector_type(4)));

// ---------------------------------------------------------------------------
// Pass 1: map[p*NCELLS + cell] = -1  (13.7 MB, wide NT stores)
// grid = NROWS/4/256 = 3348 blocks, exact cover.
// ---------------------------------------------------------------------------
__global__ __launch_bounds__(256) void pps_init_map(int* __restrict__ map) {
    const int quad = blockIdx.x * 256 + threadIdx.x;     // one 16B quad each
    v4i m1 = { -1, -1, -1, -1 };
    __builtin_nontemporal_store(m1, reinterpret_cast<v4i*>(map) + quad);
}

// ---------------------------------------------------------------------------
// Pass 2: rank scatter. atomicMax(pillar index) == last-write-wins.
// ---------------------------------------------------------------------------
__global__ __launch_bounds__(256) void pps_scatter_ids(
        const int* __restrict__ coords,   // [M,4] = (b, z, y, x)
        const int* __restrict__ gt,       // [M]
        int* __restrict__ map,
        int M) {
    const int i = blockIdx.x * 256 + threadIdx.x;
    if (i >= M) return;
    const int g = gt[i];
    if (g < 0) return;                    // background pillar -> dump row
    const int b = coords[4 * i + 0];
    const int z = coords[4 * i + 1];      // nz==1 so z==0, keep formula anyway
    const int y = coords[4 * i + 2];
    const int x = coords[4 * i + 3];
    const int row = (b * GIDS + g) * NCELLS + z + y * NXC + x;
    atomicMax(&map[row], i);
}

// ---------------------------------------------------------------------------
// Pass 3: dense gather. Block = 256 threads handles a 1024-cell spatial tile
// of plane p for ALL 64 channels. The 4KB map tile is DMA'd into LDS via the
// gfx1250 async-to-LDS path (ASYNCcnt), then each thread emits 64 coalesced
// non-temporal b128 stores (one per channel plane).
// grid = (ceil(NCELLS/1024)=210, 1, NPLANES=16)
// ---------------------------------------------------------------------------
__global__ __launch_bounds__(256) void pps_gather(
        const float* __restrict__ pf,     // [M, C]
        const int*   __restrict__ map,    // [NPLANES, NCELLS]
        float*       __restrict__ out) {  // [NPLANES, C, NCELLS]
    __shared__ int smap[1024];

    const int p     = blockIdx.z;
    const int tid   = threadIdx.x;
    const int cell0 = blockIdx.x * 1024 + tid * 4;

    // Async DMA: 16 bytes per lane, global -> LDS, no VGPR staging.
    // Clamp source so the one partial tile never reads out of bounds;
    // clamped lanes hold duplicate data that is masked off below.
    const int* srow = map + (size_t)p * NCELLS;                 // uniform -> SGPR pair
    const int  cl   = (cell0 <= NCELLS - 4) ? cell0 : (NCELLS - 4);
    const unsigned goff  = (unsigned)cl * 4u;                   // byte offset (GVS mode)
    const unsigned laddr = (unsigned)(uintptr_t)&smap[tid * 4]; // LDS byte address
    asm volatile("global_load_async_to_lds_b128 %0, %1, %2"
                 :: "v"(laddr), "v"(goff), "s"(srow) : "memory");
    asm volatile("s_wait_asynccnt 0x0" ::: "memory");
    __syncthreads();

    if (cell0 >= NCELLS) return;   // only whole quads exist (NCELLS % 4 == 0)

    const v4i idx = *reinterpret_cast<const v4i*>(&smap[tid * 4]);
    float* outp = out + ((size_t)p * CFEAT) * NCELLS + cell0;

#pragma unroll 4
    for (int c = 0; c < CFEAT; ++c) {
        v4f v;
        v.x = (idx.x >= 0) ? pf[(size_t)idx.x * CFEAT + c] : 0.0f;
        v.y = (idx.y >= 0) ? pf[(size_t)idx.y * CFEAT + c] : 0.0f;
        v.z = (idx.z >= 0) ? pf[(size_t)idx.z * CFEAT + c] : 0.0f;
        v.w = (idx.w >= 0) ? pf[(size_t)idx.w * CFEAT + c] : 0.0f;
        // 877 MB streaming write: non-temporal so the map stays L2-resident.
        __builtin_nontemporal_store(
            v, reinterpret_cast<v4f*>(outp + (size_t)c * NCELLS));
    }
}

// ---------------------------------------------------------------------------
extern "C" void kernel_launch(void* const* d_in, const int* in_sizes, int n_in,
                              void* d_out, int out_size, void* d_ws, size_t ws_size,
                              hipStream_t stream) {
    const float* pf     = (const float*)d_in[0];   // pillar_features [M, C]
    const int*   coords = (const int*)  d_in[1];   // voxel_coords   [M, 4]
    const int*   gt     = (const int*)  d_in[2];   // voxel_gt_mask  [M]
    // d_in[3] = batch_len scalar (== BATCH), constants are compiled in.
    const int M = in_sizes[2];

    float* out = (float*)d_out;
    int*   map = (int*)d_ws;                       // needs NROWS*4 = 13.7 MB

    // Pass 1: clear the rank map.
    pps_init_map<<<NROWS / 4 / 256, 256, 0, stream>>>(map);

    // Pass 2: last-write-wins rank scatter of pillar indices.
    pps_scatter_ids<<<(M + 255) / 256, 256, 0, stream>>>(coords, gt, map, M);

    // Pass 3: fully-coalesced dense gather into the output.
    dim3 grid((NCELLS + 1023) / 1024, 1, NPLANES); // (210, 1, 16)
    pps_gather<<<grid, 256, 0, stream>>>(pf, map, out);
}